// TV_76991583748615
// MI455X (gfx1250) — compile-verified
//
#include <hip/hip_runtime.h>
#include <hip/hip_bf16.h>
#include <stdint.h>

// TV-Chambolle, 2D, tau=0.25, weight=0.1  ->  tau/weight = 2.5
#define TAU   0.25f
#define SCL   2.5f
#define H     512
#define W     512

// ---- fused-kernel tile geometry --------------------------------------------
// Block = 256 threads (64x4), tile = 64 wide x 16 tall.
// LDS tiles (with halo), pitches padded so bulk copies are 16B-aligned:
//   imgS: rows gi=i0..i0+16,  cols gj=j0..j0+64   -> 17 rows, pitch 68
//   phS : rows gi=i0-1..i0+16, cols gj=j0..j0+64  -> 18 rows, pitch 68
//   pwS : rows gi=i0..i0+16,  cols gj=j0-1..j0+64 -> 17 rows, pitch 72,
//         column cc maps gj = j0-4+cc (left halo at cc=3, bulk cc=4..67,
//         right halo at cc=68) so bulk LDS offsets are 16B-aligned.
#define TILE_W 64
#define TILE_H 16
#define IMG_P 68
#define PH_P  68
#define PW_P  72
#define IMG_N (17 * IMG_P)            // 1156 (16B-aligned section)
#define PH_N  (18 * PH_P)             // 1224
#define PW_N  (17 * PW_P)             // 1224
#define LDS_N (IMG_N + PH_N + PW_N)   // 3604 floats ~ 14.4 KB

// ---- CDNA5 async global->LDS path (guarded; falls back to plain copies) ----
#if defined(__HIP_DEVICE_COMPILE__)
#if __has_builtin(__builtin_amdgcn_global_load_async_to_lds_b32)
#define HAVE_ASYNC_LDS 1
#endif
#if __has_builtin(__builtin_amdgcn_global_load_async_to_lds_b128)
#define HAVE_ASYNC_B128 1
#endif
#if __has_builtin(__builtin_amdgcn_s_wait_asynccnt)
#define HAVE_WAIT_ASYNC 1
#endif
#endif

typedef int v4i_t __attribute__((vector_size(4 * sizeof(int))));
typedef __attribute__((address_space(1))) int   gint_t;   // global (AS1)
typedef __attribute__((address_space(3))) int   lint_t;   // LDS (AS3)
typedef __attribute__((address_space(1))) v4i_t gv4i_t;   // global v4i (AS1)
typedef __attribute__((address_space(3))) v4i_t lv4i_t;   // LDS v4i (AS3)

__device__ __forceinline__ void async_copy_b32(const float* g, float* l) {
#ifdef HAVE_ASYNC_LDS
  __builtin_amdgcn_global_load_async_to_lds_b32((gint_t*)g, (lint_t*)l, 0, 0);
#else
  *l = *g;
#endif
}

__device__ __forceinline__ void async_copy_b128(const float* g, float* l) {
#ifdef HAVE_ASYNC_B128
  __builtin_amdgcn_global_load_async_to_lds_b128((gv4i_t*)g, (lv4i_t*)l, 0, 0);
#else
  *(float4*)l = *(const float4*)g;
#endif
}

__device__ __forceinline__ void wait_async_lds() {
#if defined(HAVE_ASYNC_LDS) || defined(HAVE_ASYNC_B128)
#ifdef HAVE_WAIT_ASYNC
  __builtin_amdgcn_s_wait_asynccnt(0);
#else
  asm volatile("s_wait_asynccnt 0" ::: "memory");
#endif
#endif
}

// ---------------------------------------------------------------------------
// Fused dual update:  p_out = (p_in - tau * grad(img + div(p_in))) / norm
// first!=0 => p_in treated as 0 (LDS stays zeroed; p_in never dereferenced).
// ---------------------------------------------------------------------------
__global__ __launch_bounds__(256) void tv_fused_kernel(
    const float* __restrict__ img,
    const float* __restrict__ phIn, const float* __restrict__ pwIn,
    float* __restrict__ phOut, float* __restrict__ pwOut, int first)
{
  __shared__ float sAll[LDS_N];
  float* imgS = sAll;
  float* phS  = sAll + IMG_N;
  float* pwS  = sAll + IMG_N + PH_N;

  const int tx  = threadIdx.x;           // 0..63
  const int ty  = threadIdx.y;           // 0..3
  const int tid = ty * TILE_W + tx;      // 0..255
  const int j0  = blockIdx.x * TILE_W;
  const int i0  = blockIdx.y * TILE_H;
  const size_t nbase = (size_t)blockIdx.z * ((size_t)H * W);
  const float* imgB = img + nbase;

  // Zero LDS: provides zero halo at image borders and p==0 on first iteration.
  for (int k = tid; k < LDS_N; k += 256) sAll[k] = 0.0f;
  __syncthreads();

  // ---- stage tiles with async global->LDS DMA (B128 bulk, B32 edges) ----
  // img bulk: 17 rows x 16 chunks of 4 floats
  for (int t = tid; t < 17 * 16; t += 256) {
    int row = t >> 4, ch = t & 15;
    int gi = i0 + row;
    if (gi < H)
      async_copy_b128(imgB + (size_t)gi * W + (j0 + ch * 4), imgS + row * IMG_P + ch * 4);
  }
  // img right-halo column (gj = j0+64)
  if (tid < 17) {
    int gi = i0 + tid, gj = j0 + 64;
    if (gi < H && gj < W)
      async_copy_b32(imgB + (size_t)gi * W + gj, imgS + tid * IMG_P + 64);
  }
  if (!first) {
    const float* phB = phIn + nbase;
    const float* pwB = pwIn + nbase;
    // ph bulk: 18 rows x 16 chunks (rows gi = i0-1 .. i0+16)
    for (int t = tid; t < 18 * 16; t += 256) {
      int row = t >> 4, ch = t & 15;
      int gi = i0 - 1 + row;
      if (gi >= 0 && gi < H)
        async_copy_b128(phB + (size_t)gi * W + (j0 + ch * 4), phS + row * PH_P + ch * 4);
    }
    if (tid < 18) {                       // ph right-halo column
      int gi = i0 - 1 + tid, gj = j0 + 64;
      if (gi >= 0 && gi < H && gj < W)
        async_copy_b32(phB + (size_t)gi * W + gj, phS + tid * PH_P + 64);
    }
    // pw bulk: 17 rows x 16 chunks at cc = 4..67 (gj = j0 .. j0+63)
    for (int t = tid; t < 17 * 16; t += 256) {
      int row = t >> 4, ch = t & 15;
      int gi = i0 + row;
      if (gi < H)
        async_copy_b128(pwB + (size_t)gi * W + (j0 + ch * 4), pwS + row * PW_P + 4 + ch * 4);
    }
    if (tid < 17) {                       // pw left halo (cc=3, gj=j0-1)
      int gi = i0 + tid, gj = j0 - 1;
      if (gi < H && gj >= 0)
        async_copy_b32(pwB + (size_t)gi * W + gj, pwS + tid * PW_P + 3);
    }
    if (tid >= 32 && tid < 32 + 17) {     // pw right halo (cc=68, gj=j0+64)
      int r = tid - 32;
      int gi = i0 + r, gj = j0 + 64;
      if (gi < H && gj < W)
        async_copy_b32(pwB + (size_t)gi * W + gj, pwS + r * PW_P + 68);
    }
  }
  wait_async_lds();
  __syncthreads();

  // out(i,j) = img - ph(i,j) - pw(i,j) + ph(i-1,j) + pw(i,j-1)
  // (zeroed halo rows/cols implement the boundary conditions)
  auto outv = [&](int r, int c) -> float {
    return imgS[r * IMG_P + c]
         - phS[(r + 1) * PH_P + c] - pwS[r * PW_P + (c + 4)]
         + phS[r * PH_P + c]       + pwS[r * PW_P + (c + 3)];
  };

#pragma unroll
  for (int s = 0; s < 4; ++s) {
    const int r  = ty + 4 * s;           // 0..15
    const int c  = tx;                   // 0..63
    const int gi = i0 + r, gj = j0 + c;
    const float o00 = outv(r, c);
    const float gh  = (gi < H - 1) ? (outv(r + 1, c) - o00) : 0.0f;
    const float gw  = (gj < W - 1) ? (outv(r, c + 1) - o00) : 0.0f;
    const float nrm = sqrtf(gh * gh + gw * gw) * SCL + 1.0f;
    const float p0  = phS[(r + 1) * PH_P + c];   // ph(i,j) (0 when first)
    const float q0  = pwS[r * PW_P + (c + 4)];   // pw(i,j)
    const size_t idx = nbase + (size_t)gi * W + gj;
    phOut[idx] = (p0 - TAU * gh) / nrm;
    pwOut[idx] = (q0 - TAU * gw) / nrm;
  }
}

// ---------------------------------------------------------------------------
// out = img + div(p), float4-vectorized. first!=0 => p==0 -> out = img.
// ---------------------------------------------------------------------------
__global__ __launch_bounds__(256) void tv_div_kernel(
    const float* __restrict__ img, const float* __restrict__ ph,
    const float* __restrict__ pw, float* __restrict__ out, int first)
{
  const size_t idx = ((size_t)blockIdx.x * 256 + threadIdx.x) * 4;
  const int j = (int)(idx & (W - 1));          // 0,4,...,508
  const int i = (int)((idx >> 9) & (H - 1));
  float4 v = *(const float4*)(img + idx);
  if (!first) {
    const float4 p = *(const float4*)(ph + idx);
    const float4 q = *(const float4*)(pw + idx);
    float4 pu = make_float4(0.f, 0.f, 0.f, 0.f);
    if (i > 0) pu = *(const float4*)(ph + idx - W);
    const float ql = (j > 0) ? pw[idx - 1] : q.x;  // q.x => zero contribution
    v.x += pu.x - p.x - q.x + ((j > 0) ? ql : 0.0f);
    v.y += pu.y - p.y - q.y + q.x;
    v.z += pu.z - p.z - q.z + q.y;
    v.w += pu.w - p.w - q.w + q.z;
  }
  *(float4*)(out + idx) = v;
}

// ---------------------------------------------------------------------------
// In-place dual update from precomputed out-plane (two-pass fallback path).
// float4-vectorized; neighbor predicates avoid any OOB access at buffer ends.
// ---------------------------------------------------------------------------
__global__ __launch_bounds__(256) void tv_update_kernel(
    const float* __restrict__ o, float* __restrict__ ph,
    float* __restrict__ pw, int first)
{
  const size_t idx = ((size_t)blockIdx.x * 256 + threadIdx.x) * 4;
  const int j = (int)(idx & (W - 1));
  const int i = (int)((idx >> 9) & (H - 1));
  const float4 o4 = *(const float4*)(o + idx);
  float4 od = o4;                               // => gh = 0 on last row
  if (i < H - 1) od = *(const float4*)(o + idx + W);
  float orr = o4.w;                             // => gw.w = 0 on last col
  if (j < W - 4) orr = o[idx + 4];
  const float4 gh = make_float4(od.x - o4.x, od.y - o4.y, od.z - o4.z, od.w - o4.w);
  const float4 gw = make_float4(o4.y - o4.x, o4.z - o4.y, o4.w - o4.z, orr - o4.w);
  float4 p = make_float4(0.f, 0.f, 0.f, 0.f), q = p;
  if (!first) {
    p = *(const float4*)(ph + idx);
    q = *(const float4*)(pw + idx);
  }
  float4 np, nq;
  {
    float n0 = sqrtf(gh.x * gh.x + gw.x * gw.x) * SCL + 1.0f;
    float n1 = sqrtf(gh.y * gh.y + gw.y * gw.y) * SCL + 1.0f;
    float n2 = sqrtf(gh.z * gh.z + gw.z * gw.z) * SCL + 1.0f;
    float n3 = sqrtf(gh.w * gh.w + gw.w * gw.w) * SCL + 1.0f;
    np = make_float4((p.x - TAU * gh.x) / n0, (p.y - TAU * gh.y) / n1,
                     (p.z - TAU * gh.z) / n2, (p.w - TAU * gh.w) / n3);
    nq = make_float4((q.x - TAU * gw.x) / n0, (q.y - TAU * gw.y) / n1,
                     (q.z - TAU * gw.z) / n2, (q.w - TAU * gw.w) / n3);
  }
  *(float4*)(ph + idx) = np;
  *(float4*)(pw + idx) = nq;
}

extern "C" void kernel_launch(void* const* d_in, const int* in_sizes, int n_in,
                              void* d_out, int out_size, void* d_ws, size_t ws_size,
                              hipStream_t stream) {
  (void)in_sizes; (void)n_in;
  const float* img = (const float*)d_in[0];           // x: [8,8,512,512] f32
  float* out = (float*)d_out;
  float* wsf = (float*)d_ws;

  const size_t plane      = (size_t)out_size;         // 64*512*512
  const size_t planeBytes = plane * sizeof(float);
  const unsigned nImg     = (unsigned)(plane / ((size_t)H * W));
  const int N_UPDATES     = 9;                        // 9 dual updates, then out

  dim3 fblock(TILE_W, 4, 1);
  dim3 fgrid(W / TILE_W, H / TILE_H, nImg);
  dim3 eblock(256, 1, 1);
  dim3 egrid((unsigned)(plane / 1024), 1, 1);         // float4 per thread

  if (ws_size >= 4 * planeBytes) {
    // Fused path: ping-pong (ph,pw) pairs in workspace. 10 launches total.
    float* phA = wsf;
    float* pwA = wsf + plane;
    float* phB = wsf + 2 * plane;
    float* pwB = wsf + 3 * plane;

    // update 1: p_prev == 0 (pointers passed but never dereferenced)
    tv_fused_kernel<<<fgrid, fblock, 0, stream>>>(img, phB, pwB, phA, pwA, 1);
    float *pi = phA, *qi = pwA, *po = phB, *qo = pwB;
    for (int it = 1; it < N_UPDATES; ++it) {
      tv_fused_kernel<<<fgrid, fblock, 0, stream>>>(img, pi, qi, po, qo, 0);
      float* t;
      t = pi; pi = po; po = t;
      t = qi; qi = qo; qo = t;
    }
    // out = img + div(p_9)
    tv_div_kernel<<<egrid, eblock, 0, stream>>>(img, pi, qi, out, 0);
  } else {
    // Two-pass fallback: only 2 planes of workspace; d_out doubles as the
    // per-iteration out-plane scratch until the final pass rewrites it.
    float* ph = wsf;
    float* pw = wsf + plane;
    tv_div_kernel<<<egrid, eblock, 0, stream>>>(img, ph, pw, out, 1);  // out1 = img
    tv_update_kernel<<<egrid, eblock, 0, stream>>>(out, ph, pw, 1);    // p1
    for (int it = 1; it < N_UPDATES; ++it) {
      tv_div_kernel<<<egrid, eblock, 0, stream>>>(img, ph, pw, out, 0);
      tv_update_kernel<<<egrid, eblock, 0, stream>>>(out, ph, pw, 0);
    }
    tv_div_kernel<<<egrid, eblock, 0, stream>>>(img, ph, pw, out, 0);  // out10
  }
}